// TopNLabelSmoothingCrossEntropy_64733747085573
// MI455X (gfx1250) — compile-verified
//
#include <hip/hip_runtime.h>
#include <hip/hip_bf16.h>

// TopN label-smoothing cross entropy, B = C = 8192, output = scalar mean loss.
// Memory-bound: 256MB single-use stream -> ~11us floor at 23.3 TB/s.
// Strategy: 1 block per row, NT b128 loads direct to VGPRs (single s_clause),
// fused branchless top-3 + logsumexp, wave32 shuffle butterfly + LDS cross-wave
// merge, deterministic 2-kernel reduction (no float atomics).

#define CLASS_NUM 8192
#define THREADS   256
#define V4_PER_THREAD 8   // 8 x float4 = 32 elements/thread * 256 threads = 8192

typedef float v4f __attribute__((ext_vector_type(4)));

// Branchless insert of (v,idx) into a descending sorted top-3 triple.
// 3 v_cmp + 10 v_cndmask, no exec-mask branches, dual-issue friendly.
__device__ __forceinline__ void ins3(float v, int idx,
                                     float& v0, int& i0,
                                     float& v1, int& i1,
                                     float& v2, int& i2) {
    const bool g0 = v > v0;
    const bool g1 = v > v1;
    const bool g2 = v > v2;
    const float nv2 = g1 ? v1 : (g2 ? v   : v2);
    const int   ni2 = g1 ? i1 : (g2 ? idx : i2);
    const float nv1 = g0 ? v0 : (g1 ? v   : v1);
    const int   ni1 = g0 ? i0 : (g1 ? idx : i1);
    const float nv0 = g0 ? v   : v0;
    const int   ni0 = g0 ? idx : i0;
    v0 = nv0; i0 = ni0;
    v1 = nv1; i1 = ni1;
    v2 = nv2; i2 = ni2;
}

__global__ __launch_bounds__(THREADS)
void topn_ls_ce_row_kernel(const float* __restrict__ preds,
                           const long long* __restrict__ targets,
                           float* __restrict__ row_loss) {
    const int C   = CLASS_NUM;
    const int row = blockIdx.x;
    const int tid = threadIdx.x;

    const v4f* rowp = (const v4f*)(preds + (size_t)row * (size_t)C);

    // Issue all 8 non-temporal 128b loads up front (coalesced, streaming).
    v4f x[V4_PER_THREAD];
#pragma unroll
    for (int k = 0; k < V4_PER_THREAD; ++k)
        x[k] = __builtin_nontemporal_load(rowp + (k * THREADS + tid));

    // Pass 1: per-thread top-3 (v0 is also the per-thread max).
    float v0 = -INFINITY, v1 = -INFINITY, v2 = -INFINITY;
    int   i0 = -1, i1 = -1, i2 = -1;
#pragma unroll
    for (int k = 0; k < V4_PER_THREAD; ++k) {
        const int col = (k * THREADS + tid) * 4;
#pragma unroll
        for (int c = 0; c < 4; ++c)
            ins3(x[k][c], col + c, v0, i0, v1, i1, v2, i2);
    }
    const float m = v0;

    // Pass 2: sum of exp(x - m) over the 32 register-resident elements.
    float s = 0.0f;
#pragma unroll
    for (int k = 0; k < V4_PER_THREAD; ++k)
#pragma unroll
        for (int c = 0; c < 4; ++c)
            s += __expf(x[k][c] - m);

    // Wave32 butterfly: merge (m,s) online-softmax style + merge top-3 triples.
    float mm = m;
#pragma unroll
    for (int off = 16; off > 0; off >>= 1) {
        float mo = __shfl_xor(mm, off, 32);
        float so = __shfl_xor(s,  off, 32);
        float w0 = __shfl_xor(v0, off, 32); int j0 = __shfl_xor(i0, off, 32);
        float w1 = __shfl_xor(v1, off, 32); int j1 = __shfl_xor(i1, off, 32);
        float w2 = __shfl_xor(v2, off, 32); int j2 = __shfl_xor(i2, off, 32);
        float nm = fmaxf(mm, mo);
        s = s * __expf(mm - nm) + so * __expf(mo - nm);
        mm = nm;
        ins3(w0, j0, v0, i0, v1, i1, v2, i2);
        ins3(w1, j1, v0, i0, v1, i1, v2, i2);
        ins3(w2, j2, v0, i0, v1, i1, v2, i2);
    }

    // Cross-wave merge via LDS (8 waves per block).
    __shared__ float sm[8], ss[8], sv[8][3];
    __shared__ int   si[8][3];
    const int wid  = tid >> 5;
    const int lane = tid & 31;
    if (lane == 0) {
        sm[wid] = mm; ss[wid] = s;
        sv[wid][0] = v0; sv[wid][1] = v1; sv[wid][2] = v2;
        si[wid][0] = i0; si[wid][1] = i1; si[wid][2] = i2;
    }
    __syncthreads();

    if (tid == 0) {
        for (int w = 1; w < 8; ++w) {
            float mo = sm[w], so = ss[w];
            float nm = fmaxf(mm, mo);
            s = s * __expf(mm - nm) + so * __expf(mo - nm);
            mm = nm;
            ins3(sv[w][0], si[w][0], v0, i0, v1, i1, v2, i2);
            ins3(sv[w][1], si[w][1], v0, i0, v1, i1, v2, i2);
            ins3(sv[w][2], si[w][2], v0, i0, v1, i1, v2, i2);
        }
        const float logZ = mm + __logf(s);

        // Sparse target entries. targets is int64 in the reference.
        const int   tcol = (int)targets[row];
        const float dval = preds[(size_t)row * (size_t)C + (size_t)row];
        const float tval = preds[(size_t)row * (size_t)C + (size_t)tcol];

        // Rank of class `row` among top-3 (3 if absent).
        const int r  = (i0 == row) ? 0 : (i1 == row) ? 1 : (i2 == row) ? 2 : 3;
        // p_j = (j-1) + ((j-1) >= r) for j = 1,2  (skip the ground-truth slot).
        const int p0 = (r == 0) ? 1 : 0;
        const int p1 = (r <= 1) ? 2 : 1;

        const float tkv[3] = { v0, v1, v2 };
        const int   tki[3] = { i0, i1, i2 };

        // In-order scatter with overwrite semantics: later .set() wins.
        int   cls[4]  = { tcol, row, tki[p0], tki[p1] };
        float wts[4]  = { 1.0f, 0.7f, 0.2f, 0.1f };
        float vals[4] = { tval, dval, tkv[p0], tkv[p1] };
#pragma unroll
        for (int a = 0; a < 4; ++a)
#pragma unroll
            for (int b = a + 1; b < 4; ++b)
                if (cls[a] == cls[b]) wts[a] = 0.0f;

        float wsum = 0.0f, dot = 0.0f;
#pragma unroll
        for (int a = 0; a < 4; ++a) { wsum += wts[a]; dot += wts[a] * vals[a]; }

        // loss_i = -sum w*(pred - logZ) = wsum*logZ - dot
        row_loss[row] = wsum * logZ - dot;
    }
}

// Deterministic final mean: fixed sequential order per thread + LDS tree.
__global__ __launch_bounds__(THREADS)
void topn_ls_ce_reduce_kernel(const float* __restrict__ row_loss,
                              float* __restrict__ out) {
    __shared__ float acc[THREADS];
    const int tid = threadIdx.x;
    float a = 0.0f;
    for (int j = 0; j < CLASS_NUM / THREADS; ++j)
        a += row_loss[tid + THREADS * j];
    acc[tid] = a;
    __syncthreads();
#pragma unroll
    for (int off = THREADS / 2; off > 0; off >>= 1) {
        if (tid < off) acc[tid] += acc[tid + off];
        __syncthreads();
    }
    if (tid == 0) out[0] = acc[0] * (1.0f / (float)CLASS_NUM);
}

extern "C" void kernel_launch(void* const* d_in, const int* in_sizes, int n_in,
                              void* d_out, int out_size, void* d_ws, size_t ws_size,
                              hipStream_t stream) {
    (void)in_sizes; (void)n_in; (void)out_size; (void)ws_size;
    const float*     preds   = (const float*)d_in[0];
    const long long* targets = (const long long*)d_in[1];   // int64 targets
    float*           out     = (float*)d_out;
    float*           rloss   = (float*)d_ws;                // 8192 floats of scratch

    topn_ls_ce_row_kernel<<<CLASS_NUM, THREADS, 0, stream>>>(preds, targets, rloss);
    topn_ls_ce_reduce_kernel<<<1, THREADS, 0, stream>>>(rloss, out);
}